// SimplePitchEstimator_29515015258337
// MI455X (gfx1250) — compile-verified
//
#include <hip/hip_runtime.h>

typedef float v2f __attribute__((ext_vector_type(2)));
typedef float v8f __attribute__((ext_vector_type(8)));

#define FRAME_LEN 160
#define HOP 256
#define MIN_LAG 40
#define WAVES_PER_BLOCK 8
#define XP_SIZE 416   // 240 zeros + 160 samples (+pad); reads hit [0,399]
#define XR_SIZE 176   // 160 reversed samples + 16 zeros; reads hit [0,174]
#define NEG_INF (-3.402823466e+38f)

// One wave (32 lanes) computes one frame's 120 self-convolution values as a
// single 16x16 f32 WMMA tile accumulated over K=160 (40x V_WMMA_F32_16X16X4_F32):
//   D[q,p] = sum_k xr[k+q] * x[k-16p]  =  conv[159 - (16p+q)]
__global__ __launch_bounds__(256) void pitch_wmma_kernel(
    const float* __restrict__ audio, float* __restrict__ out,
    int S, int T, int nframes) {
  __shared__ float s_xp[WAVES_PER_BLOCK][XP_SIZE];
  __shared__ float s_xr[WAVES_PER_BLOCK][XR_SIZE];

  const int lane = threadIdx.x & 31;
  const int wave = threadIdx.x >> 5;
  int f = blockIdx.x * WAVES_PER_BLOCK + wave;
  if (f >= nframes) f = nframes - 1;  // clamp (duplicate work) to keep EXEC all-1s for WMMA
  const int b = f / T;
  const int t = f - b * T;
  const float* __restrict__ src = audio + (size_t)b * (size_t)S + (size_t)t * HOP;

  // ---- load 160 samples (5 per lane), mean-center, silent check ----
  float x0 = src[lane];
  float x1 = src[lane + 32];
  float x2 = src[lane + 64];
  float x3 = src[lane + 96];
  float x4 = src[lane + 128];

  float sum = x0 + x1 + x2 + x3 + x4;
#pragma unroll
  for (int off = 16; off; off >>= 1) sum += __shfl_xor(sum, off, 32);
  const float mean = sum * (1.0f / 160.0f);
  x0 -= mean; x1 -= mean; x2 -= mean; x3 -= mean; x4 -= mean;

  float mx = fmaxf(fmaxf(fmaxf(fabsf(x0), fabsf(x1)),
                         fmaxf(fabsf(x2), fabsf(x3))), fabsf(x4));
#pragma unroll
  for (int off = 16; off; off >>= 1) mx = fmaxf(mx, __shfl_xor(mx, off, 32));
  const bool silent = mx < 1e-8f;

  // ---- stage zero-padded x and reversed xr into LDS ----
  float* xp = s_xp[wave];
  float* xr = s_xr[wave];
  for (int j = lane; j < 240; j += 32) xp[j] = 0.0f;             // left zero pad
  if (lane < XR_SIZE - FRAME_LEN) xr[FRAME_LEN + lane] = 0.0f;   // xr tail zeros
  xp[240 + lane      ] = x0;  xr[159 - lane      ] = x0;
  xp[240 + lane + 32 ] = x1;  xr[159 - lane - 32 ] = x1;
  xp[240 + lane + 64 ] = x2;  xr[159 - lane - 64 ] = x2;
  xp[240 + lane + 96 ] = x3;  xr[159 - lane - 96 ] = x3;
  xp[240 + lane + 128] = x4;  xr[159 - lane - 128] = x4;
  __syncthreads();

  // ---- K-loop: 40x V_WMMA_F32_16X16X4_F32, fully unrolled ----
  const int n  = lane & 15;  // A row q / B column p
  const int hf = lane >> 4;  // lane-half selects K pair {0,1} vs {2,3}
  const float* __restrict__ xra = xr + 2 * hf + n;
  const float* __restrict__ xpb = xp + 240 + 2 * hf - 16 * n;
  v8f c = {};
#pragma unroll
  for (int k0 = 0; k0 < FRAME_LEN; k0 += 4) {
    v2f a, bb;
    a.x  = xra[k0];           // A[q=n, K=k0+2*hf]   = xr[K+q]
    a.y  = xra[k0 + 1];       // A[q=n, K=k0+2*hf+1]
    bb.x = xpb[k0];           // B[K=k0+2*hf,   p=n] = x[K-16p]
    bb.y = xpb[k0 + 1];       // B[K=k0+2*hf+1, p=n]
    c = __builtin_amdgcn_wmma_f32_16x16x4_f32(
        /*neg_a=*/false, a, /*neg_b=*/false, bb,
        /*c_mod=*/(short)0, c, /*reuse_a=*/false, /*reuse_b=*/false);
  }

  // ---- branchless argmax over valid lags ----
  // lane holds D[v + 8*hf, n] for v=0..7; lag L = 16n + v + 8hf, conv idx k = 159-L
  union { v8f v; float f[8]; } cu; cu.v = c;
  float bestv = NEG_INF;
  int   bestk = 1 << 30;
#pragma unroll
  for (int v = 0; v < 8; ++v) {
    const int m = v + 8 * hf;
    const int L = 16 * n + m;
    const int k = 159 - L;
    const float val = (L <= 159 - MIN_LAG) ? cu.f[v] : NEG_INF;
    const bool better = (val > bestv) | ((val == bestv) & (k < bestk));
    bestv = better ? val : bestv;
    bestk = better ? k : bestk;
  }
#pragma unroll
  for (int off = 16; off; off >>= 1) {
    const float ov = __shfl_xor(bestv, off, 32);
    const int   ok = __shfl_xor(bestk, off, 32);
    const bool better = (ov > bestv) | ((ov == bestv) & (ok < bestk));
    bestv = better ? ov : bestv;
    bestk = better ? ok : bestk;
  }

  if (lane == 0) {
    out[f] = silent ? 0.0f : 16000.0f / (float)bestk;
  }
}

extern "C" void kernel_launch(void* const* d_in, const int* in_sizes, int n_in,
                              void* d_out, int out_size, void* d_ws, size_t ws_size,
                              hipStream_t stream) {
  (void)n_in; (void)d_ws; (void)ws_size;
  const float* audio = (const float*)d_in[0];
  float* out = (float*)d_out;
  const int B = 16;
  const int S = in_sizes[0] / B;                       // 1048576
  const int T = (S - FRAME_LEN) / HOP + 1;             // 4096
  const int nframes = B * T;                           // == out_size
  (void)out_size;
  const int blocks = (nframes + WAVES_PER_BLOCK - 1) / WAVES_PER_BLOCK;
  pitch_wmma_kernel<<<blocks, 32 * WAVES_PER_BLOCK, 0, stream>>>(
      audio, out, S, T, nframes);
}